// Ag3ChargeStateModel_52158082842765
// MI455X (gfx1250) — compile-verified
//
#include <hip/hip_runtime.h>
#include <hip/hip_bf16.h>

typedef float v2f __attribute__((ext_vector_type(2)));
typedef float v8f __attribute__((ext_vector_type(8)));

#define NATOMS 2048
#define NRBF   16
#define NHID   32
#define NTILES (NATOMS / 16)
#define WAVES  8          // waves per block in kernel 1

static constexpr float CUT = 5.0f;
static constexpr float L2E = 1.4426950408889634f;   // log2(e)

// Ratio constants for the Gaussian ladder: DR[k] = exp(-(2k-1)), so
// g_k = g_{k-1} * exp(6d) * DR[k]   (exact for centers c_k = k/3, gamma = 9).
__device__ __constant__ const float DR[16] = {
    0.0f,                      // unused (k=0 is an anchor)
    0.36787944117144233f,      // e^-1
    0.04978706836786394f,      // e^-3
    0.006737946999085467f,     // e^-5
    0.0009118819655545162f,    // e^-7
    0.00012340980408667956f,   // e^-9
    1.670170079024566e-05f,    // e^-11
    2.2603294069810542e-06f,   // e^-13
    0.0f,                      // unused (k=8 is an anchor)
    4.139937718785167e-08f,    // e^-17
    5.602796437537268e-09f,    // e^-19
    7.582560427911907e-10f,    // e^-21
    1.026187963170189e-10f,    // e^-23
    1.3887943864964021e-11f,   // e^-25
    1.8795288165390832e-12f,   // e^-27
    2.543665647376923e-13f     // e^-29
};

// ---------------------------------------------------------------------------
// Kernel 1: masked RBF neighbor-sum features via WMMA Gram matrix.
// Homogeneous trick: A-row = (-2x_i, -2y_i, -2z_i, 1), B-col = (x_j, y_j, z_j, |pj|^2)
//   => C[m][n] = |pj|^2 - 2*dot(pi,pj);  d2 = |pi|^2 + C  (one WMMA, no extra LDS)
// One block (8 waves) per 16-row i-strip; wave w handles j-tiles w, w+8, ...
// RBF ladder: 3 exps per pair (anchors k=0 and k=8, plus E = exp(6d)) and two
// muls per chained center, instead of 16 exps.
// ---------------------------------------------------------------------------
__global__ __launch_bounds__(256) void rbf_feat_kernel(const float* __restrict__ pos,
                                                       float* __restrict__ feat) {
  __shared__ float s_dot[WAVES][16][16];  // per-wave 16x16 (sqj - 2*dot) tile
  __shared__ float s_feat[16][16];        // per-strip feature accumulator

  const int lane = threadIdx.x & 31;
  const int wv   = threadIdx.x >> 5;      // 0..7
  const int i0   = blockIdx.x << 4;       // one strip per block
  const int n    = lane & 15;
  const int hf   = lane >> 4;

  // zero the shared feature accumulator (256 threads cover 16x16 entries)
  s_feat[threadIdx.x >> 4][threadIdx.x & 15] = 0.0f;

  // A fragment (16x4 f32): lanes 0-15 hold K=0,1; lanes 16-31 hold K=2,3
  //   A[m][:] = (-2x, -2y, -2z, 1)
  const int   irow = i0 + n;
  const float ax = pos[irow * 3 + 0];
  const float ay = pos[irow * 3 + 1];
  const float az = pos[irow * 3 + 2];
  v2f afrag;
  afrag.x = hf ? (-2.0f * az) : (-2.0f * ax);
  afrag.y = hf ? 1.0f         : (-2.0f * ay);
  const float sq_i = ax * ax + ay * ay + az * az;  // this lane's processing row is i0+n

  const float width = CUT / (float)(NRBF - 1);          // 1/3
  const float g2    = (-1.0f / (width * width)) * L2E;  // -9*log2(e), exp2 domain
  const float e6    = (2.0f / width) * L2E;             // 6*log2(e),  exp2 domain

  float facc[NRBF];
#pragma unroll
  for (int k = 0; k < NRBF; ++k) facc[k] = 0.0f;

  for (int jt = wv; jt < NTILES; jt += WAVES) {
    const int   j0   = jt << 4;
    const int   jrow = j0 + n;
    const float bx = pos[jrow * 3 + 0];
    const float by = pos[jrow * 3 + 1];
    const float bz = pos[jrow * 3 + 2];
    // B fragment (4x16 f32): lanes 0-15 hold K=0,1; lanes 16-31 hold K=2,3; N = lane&15
    //   B[:][n] = (x, y, z, |pj|^2)
    v2f bfrag;
    bfrag.x = hf ? bz : bx;
    bfrag.y = hf ? (bx * bx + by * by + bz * bz) : by;

    v8f c = {};
    c = __builtin_amdgcn_wmma_f32_16x16x4_f32(false, afrag, false, bfrag,
                                              (short)0, c, false, false);

    // C layout: VGPR r, lane l -> (M = r + 8*(l>>4), N = l&15)
#pragma unroll
    for (int r = 0; r < 8; ++r)
      s_dot[wv][r + (hf << 3)][n] = c[r];

    // Intra-wave store->load ordering: LDS ops are in-order per wave; this
    // blocks compiler reordering and drains DScnt.
    asm volatile("s_wait_dscnt 0x0" ::: "memory");

    // Re-distributed ownership: lane owns row i = i0+n, columns j0+8*hf .. +7
    const int jbase = j0 + (hf << 3);
#pragma unroll
    for (int u = 0; u < 8; ++u) {
      const float d2 = sq_i + s_dot[wv][n][(hf << 3) + u];  // sqi + sqj - 2 dot
      const int   j  = jbase + u;
      if ((j != irow) && (d2 > 0.0f) && (d2 < CUT * CUT)) {
        const float dist = __builtin_amdgcn_sqrtf(d2);      // d in (0,5)
        const float E    = __builtin_amdgcn_exp2f(e6 * dist);  // exp(6d) <= e^30, fits f32
        // anchor k=0: exp(-9 d^2)
        float g = __builtin_amdgcn_exp2f(g2 * d2);
        facc[0] += g;
#pragma unroll
        for (int k = 1; k < 8; ++k) {       // ladder k=1..7
          g = (g * E) * DR[k];              // intermediates <= e^29
          facc[k] += g;
        }
        // re-anchor k=8: exp(-9 (d - 8/3)^2)  (guards against g0 underflow)
        const float t8 = dist - 8.0f * width;
        g = __builtin_amdgcn_exp2f(g2 * t8 * t8);
        facc[8] += g;
#pragma unroll
        for (int k = 9; k < 16; ++k) {      // ladder k=9..15
          g = (g * E) * DR[k];
          facc[k] += g;
        }
      }
    }
    // keep next iteration's LDS stores after this iteration's loads
    asm volatile("" ::: "memory");
  }

  __syncthreads();  // s_feat zeroed + all waves done accumulating
#pragma unroll
  for (int k = 0; k < NRBF; ++k)
    atomicAdd(&s_feat[n][k], facc[k]);   // ds_add_f32
  __syncthreads();

  // 256 threads write the 16x16 strip features
  const int row = threadIdx.x >> 4;
  const int kk  = threadIdx.x & 15;
  feat[(i0 + row) * NRBF + kk] = s_feat[row][kk];
}

// ---------------------------------------------------------------------------
// Kernel 2: [feat | cs_embed] @ W1 (K=48, via 12 WMMA k-steps x 2 n-tiles),
// bias + SiLU, dot with W2, global scalar reduction (+ 2048*b2 once).
// ---------------------------------------------------------------------------
__global__ __launch_bounds__(128) void mlp_reduce_kernel(const float* __restrict__ feat,
                                                         const int*   __restrict__ charge,
                                                         const float* __restrict__ emb,  // [2,32]
                                                         const float* __restrict__ W1,   // [48,32]
                                                         const float* __restrict__ b1,   // [32]
                                                         const float* __restrict__ W2,   // [32]
                                                         const float* __restrict__ b2,   // [1]
                                                         float* __restrict__ out) {
  __shared__ float s_sum;
  if (threadIdx.x == 0) s_sum = 0.0f;
  __syncthreads();

  const int lane  = threadIdx.x & 31;
  const int wv    = threadIdx.x >> 5;
  const int strip = (blockIdx.x << 2) + wv;
  const int i0    = strip << 4;
  const int n     = lane & 15;
  const int hf    = lane >> 4;

  const int    cs  = (charge[0] < 0) ? 0 : 1;
  const float* cse = emb + cs * NHID;

  v8f c0 = {};  // hidden units 0..15
  v8f c1 = {};  // hidden units 16..31
#pragma unroll
  for (int kb = 0; kb < NRBF + NHID; kb += 4) {
    const int k0 = kb + (hf << 1);
    const int k1 = k0 + 1;
    // A fragment row = i0+n; combined[k] = feat[:,k] for k<16 else cs_embed[k-16]
    const float a0 = (k0 < NRBF) ? feat[(i0 + n) * NRBF + k0] : cse[k0 - NRBF];
    const float a1 = (k1 < NRBF) ? feat[(i0 + n) * NRBF + k1] : cse[k1 - NRBF];
    v2f a = {a0, a1};
    v2f blo = {W1[k0 * NHID + n],      W1[k1 * NHID + n]};
    v2f bhi = {W1[k0 * NHID + 16 + n], W1[k1 * NHID + 16 + n]};
    c0 = __builtin_amdgcn_wmma_f32_16x16x4_f32(false, a, false, blo, (short)0, c0, false, false);
    c1 = __builtin_amdgcn_wmma_f32_16x16x4_f32(false, a, false, bhi, (short)0, c1, false, false);
  }

  // C element (r, lane) -> atom i0 + r + 8*hf, hidden unit n (c0) / 16+n (c1).
  // Output is the sum over atoms of silu(h) @ W2, so weight every element by W2[col].
  float t = 0.0f;
#pragma unroll
  for (int r = 0; r < 8; ++r) {
    {
      const float h = c0[r] + b1[n];
      const float sig = __builtin_amdgcn_rcpf(1.0f + __builtin_amdgcn_exp2f(-L2E * h));
      t += h * sig * W2[n];
    }
    {
      const float h = c1[r] + b1[16 + n];
      const float sig = __builtin_amdgcn_rcpf(1.0f + __builtin_amdgcn_exp2f(-L2E * h));
      t += h * sig * W2[16 + n];
    }
  }
  if (blockIdx.x == 0 && threadIdx.x == 0) t += (float)NATOMS * b2[0];

  atomicAdd(&s_sum, t);
  __syncthreads();
  if (threadIdx.x == 0) atomicAdd(out, s_sum);
}

__global__ void zero_out_kernel(float* __restrict__ out) { out[0] = 0.0f; }

// ---------------------------------------------------------------------------
extern "C" void kernel_launch(void* const* d_in, const int* in_sizes, int n_in,
                              void* d_out, int out_size, void* d_ws, size_t ws_size,
                              hipStream_t stream) {
  const float* pos    = (const float*)d_in[0];  // [2048,3]
  const int*   charge = (const int*)  d_in[1];  // scalar
  const float* emb    = (const float*)d_in[2];  // [2,32]
  const float* W1     = (const float*)d_in[3];  // [48,32]
  const float* b1     = (const float*)d_in[4];  // [32]
  const float* W2     = (const float*)d_in[5];  // [32,1]
  const float* b2     = (const float*)d_in[6];  // [1]
  float*       out    = (float*)d_out;
  float*       feat   = (float*)d_ws;           // [2048,16] = 128 KB scratch

  (void)in_sizes; (void)n_in; (void)out_size; (void)ws_size;

  zero_out_kernel<<<1, 1, 0, stream>>>(out);
  rbf_feat_kernel<<<NTILES, 256, 0, stream>>>(pos, feat);        // 128 blocks, 1024 waves
  mlp_reduce_kernel<<<NTILES / 4, 128, 0, stream>>>(feat, charge, emb, W1, b1, W2, b2, out);
}